// Solvatation_54803782697320
// MI455X (gfx1250) — compile-verified
//
#include <hip/hip_runtime.h>
#include <cmath>

// ---------------------------------------------------------------------------
// Solvatation scatter-reduce, MI455X (gfx1250, wave32).
// Memory-bound streaming kernel: 128 B/atom in, atomic f32 scatter to a
// 717 KB L2-resident table. No matmul structure -> no WMMA by design.
// ---------------------------------------------------------------------------

#define PADV (-999)

constexpr int NTYPES_C = 40;
constexpr int NFAKE_C  = 4;
constexpr int NALTER_C = 4;
constexpr int B_C = 8, NCH_C = 4, NRES_C = 700;
constexpr int TABLE_C = B_C * NCH_C * NRES_C * NALTER_C;   // 89600 per output

__global__ void zero_out_kernel(float* __restrict__ out, int n) {
    int i = blockIdx.x * blockDim.x + threadIdx.x;
    if (i < n) out[i] = 0.0f;
}

__launch_bounds__(256)
__global__ void solv_kernel(const int4*   __restrict__ desc,        // (N,4) i32
                            const float4* __restrict__ facc,        // (N,4) f32
                            const float4* __restrict__ contRat,     // (N,4) f32
                            const float4* __restrict__ contRatPol,  // (N,4,4) f32 -> 4x float4/atom
                            const int4*   __restrict__ hbonds,      // (N,4) i32
                            const float*  __restrict__ props,       // (40,5) f32
                            const int*    __restrict__ fakeRot,     // (40,4) i32
                            const float*  __restrict__ wH,
                            const float*  __restrict__ wP,
                            const float*  __restrict__ wPW,
                            float* __restrict__ outP,
                            float* __restrict__ outH,
                            int nAtoms, float dcp, float ionScale)
{
    // ---- per-type derived tables in LDS (computed once per block) ----------
    __shared__ float sOcc[NTYPES_C], sInvD[NTYPES_C], sHCoef[NTYPES_C],
                     sScale[NTYPES_C], sNHE[NTYPES_C];
    __shared__ int   sFlags[NTYPES_C], sMasks[NTYPES_C];

    if (threadIdx.x < NTYPES_C) {
        const int   t      = threadIdx.x;
        const float ener   = props[t * 5 + 0];
        const float occ    = props[t * 5 + 1];
        const float occmax = props[t * 5 + 2];
        const float virt   = props[t * 5 + 3];
        const float chg    = props[t * 5 + 4];

        const bool charged_ok = (chg != (float)PADV);
        const bool zero_denom = (occmax != occ);
        const bool is_nit  = (t == 0) || (t == 9);
        const bool is_acc  = (t == 2) || (t == 4) || (t == 6) || (t == 8);
        const bool lin     = charged_ok && !is_nit && is_acc;
        const bool nonvirt = (virt == 0.0f);
        const bool hydro    = (ener <= 0.0f) && nonvirt;
        const bool nonhydro = (ener >= 0.0f) && nonvirt;

        int m0 = 0, m1 = 0, m2 = 0;
        #pragma unroll
        for (int j = 0; j < NFAKE_C; ++j) {
            const int r = fakeRot[t * NFAKE_C + j];
            if (r == 0) m0 |= 1 << j;
            if (r == 1) m1 |= 1 << j;
            if (r == 2) m2 |= 1 << j;
        }
        sOcc[t]   = occ;
        sInvD[t]  = zero_denom ? (1.0f / (occmax - occ)) : 1.0f;
        sHCoef[t] = hydro ? (ener - dcp) * ionScale : 0.0f;   // folded hydro gate
        sScale[t] = lin ? 1.0f : 2.0f;
        sNHE[t]   = nonhydro ? ener : 0.0f;                   // folded nonhydro gate
        sFlags[t] = (charged_ok && zero_denom) ? 1 : 0;
        sMasks[t] = m0 | (m1 << 4) | (m2 << 8);
    }
    __syncthreads();

    // scalar weights -> coefficients (uniform, trivially cheap)
    const float cH  = 1.0f - tanhf(-wH[0]);
    const float cP  = 1.0f - tanhf(-wP[0]);
    const float cPW = 1.0f - tanhf(-wPW[0]);

    const int stride = gridDim.x * blockDim.x;
    for (int i = blockIdx.x * blockDim.x + threadIdx.x; i < nAtoms; i += stride) {
        // stream prefetch one grid-stride ahead (global_prefetch_b8)
        if (i + stride < nAtoms) {
            __builtin_prefetch(&contRatPol[(size_t)(i + stride) * 4], 0, 0);
            __builtin_prefetch(&facc[i + stride], 0, 0);
        }

        const int4 d = desc[i];
        if (d.w == PADV) continue;                 // invalid atoms add exactly 0
        const int t = d.x;

        const float4 fa = facc[i];
        const float4 cr = contRat[i];
        const int4   hb = hbonds[i];
        const float4 pA0 = contRatPol[(size_t)i * 4 + 0];
        const float4 pA1 = contRatPol[(size_t)i * 4 + 1];
        const float4 pA2 = contRatPol[(size_t)i * 4 + 2];
        const float4 pA3 = contRatPol[(size_t)i * 4 + 3];

        const float occ   = sOcc[t];
        const float invD  = sInvD[t];
        const float scale = sScale[t];
        const float nhe   = sNHE[t];
        const float hc    = cH * sHCoef[t];        // 0 unless hydro type
        const int   ok    = sFlags[t];             // charged_ok && zero_denom
        const int   masks = sMasks[t];
        const int   m0 = masks & 0xF, m1 = (masks >> 4) & 0xF, m2 = (masks >> 8) & 0xF;

        const int base = ((d.y * NCH_C + d.z) * NRES_C + d.w) * NALTER_C;

        const float  faA[4] = {fa.x, fa.y, fa.z, fa.w};
        const float  crA[4] = {cr.x, cr.y, cr.z, cr.w};
        const int    hbA[4] = {hb.x, hb.y, hb.z, hb.w};
        const float4 pA[4]  = {pA0, pA1, pA2, pA3};

        const bool doH = (hc != 0.0f);

        #pragma unroll
        for (int a = 0; a < NALTER_C; ++a) {
            const float4 p = pA[a];
            // gtMask: which fakes have contRatPol > 0 for this alternate
            const int gt = (p.x > 0.0f ? 1 : 0) | (p.y > 0.0f ? 2 : 0) |
                           (p.z > 0.0f ? 4 : 0) | (p.w > 0.0f ? 8 : 0);
            // all_fake_occ over rotamer classes k=0..2
            const bool afo = (m0 && !(m0 & ~gt)) ||
                             (m1 && !(m1 & ~gt)) ||
                             (m2 && !(m2 & ~gt));
            // solvPCorr (the existingFake&crp==0 term is redundant once gt!=0)
            const bool spc = (gt != 0) && ok && afo && (hbA[a] == 0);

            float sp;
            if (spc) {
                const float use = scale * (p.x + p.y + p.z + p.w) + crA[a];
                const float val = fminf(fmaxf((use - occ) * invD, 1e-8f), 2.0f);
                sp = cPW * val * nhe;              // m1 branch, * enerG (folded in nhe)
            } else {
                sp = cP * faA[a] * nhe;            // m2 branch, * enerG
            }
            if (sp != 0.0f) unsafeAtomicAdd(&outP[base + a], sp);
            if (doH)        unsafeAtomicAdd(&outH[base + a], hc * faA[a]);
        }
    }
}

extern "C" void kernel_launch(void* const* d_in, const int* in_sizes, int n_in,
                              void* d_out, int out_size, void* d_ws, size_t ws_size,
                              hipStream_t stream) {
    const int nAtoms = in_sizes[0] / 4;

    const int4*   desc       = (const int4*)  d_in[0];
    const float4* facc       = (const float4*)d_in[1];
    const float4* contRat    = (const float4*)d_in[2];
    const float4* contRatPol = (const float4*)d_in[3];
    const int4*   hbonds     = (const int4*)  d_in[4];
    // d_in[5] = atomDisul, unused by the reference -> never touched (saves 32 MB)
    const float*  props      = (const float*) d_in[6];
    const int*    fakeRot    = (const int*)   d_in[7];
    const float*  wH         = (const float*) d_in[8];
    const float*  wP         = (const float*) d_in[9];
    const float*  wPW        = (const float*) d_in[10];

    float* outP = (float*)d_out;
    float* outH = outP + TABLE_C;

    // Host-side constants (match reference exactly)
    const double dcp_d = (0.008 - 5e-05 * (298.0 - 273.0)) * std::log(298.0 / 273.0);
    const double corr  = std::sqrt(0.05) / 3.9;

    // Outputs are accumulated with atomics -> must re-zero on every launch
    zero_out_kernel<<<(2 * TABLE_C + 255) / 256, 256, 0, stream>>>(outP, 2 * TABLE_C);

    const int threads = 256;                       // 8 waves (wave32) per block
    int blocks = (nAtoms + threads * 4 - 1) / (threads * 4);   // ~4 atoms/thread
    if (blocks < 1) blocks = 1;
    solv_kernel<<<blocks, threads, 0, stream>>>(desc, facc, contRat, contRatPol,
                                                hbonds, props, fakeRot,
                                                wH, wP, wPW,
                                                outP, outH, nAtoms,
                                                (float)dcp_d, (float)(1.0 + corr));
}